// TverskyCrossEntropyDiceWeightedLoss_2765958938861
// MI455X (gfx1250) — compile-verified
//
#include <hip/hip_runtime.h>

typedef __attribute__((ext_vector_type(2))) float v2f;
typedef __attribute__((ext_vector_type(8))) float v8f;

#define NC   21
#define BB   8
#define HH   512
#define WW   512
#define HW   (HH * WW)          // 262144
#define CHW  (NC * HW)          // 5505024
#define NROWS (BB * HH)         // 4096
#define CT   32                 // columns per wave tile
#define NTILE (WW / CT)         // 16 column tiles
#define ROWCH 16                // grid.y
#define ROWS_PER_BLOCK (NROWS / ROWCH)          // 256
#define P1_THREADS 512
#define P1_WAVES   (P1_THREADS / 32)            // 16
#define ROWS_PER_WAVE (ROWS_PER_BLOCK / P1_WAVES)  // 16
#define EPSF 1e-6f
#define NCW  (NC * WW)          // 10752
#define FW_WAVES 8              // final-kernel waves (21 * 8 * 64 = 10752 exactly)

// Workspace layout (floats): S, inter, T, nll, pcnt, match — each [NC][WW]
#define WS_S  0
#define WS_I  (1 * NCW)
#define WS_T  (2 * NCW)
#define WS_N  (3 * NCW)
#define WS_P  (4 * NCW)
#define WS_M  (5 * NCW)
#define WS_FLOATS (6 * NCW)

__global__ __launch_bounds__(P1_THREADS) void tcedw_pass1(
    const float* __restrict__ pred, const int* __restrict__ target, float* __restrict__ ws)
{
    __shared__ float sS[NC * CT], sI[NC * CT], sT[NC * CT];
    __shared__ float sN[NC * CT], sP[NC * CT], sM[NC * CT];

    const int tid = threadIdx.x;
    for (int i = tid; i < NC * CT; i += P1_THREADS) {
        sS[i] = 0.f; sI[i] = 0.f; sT[i] = 0.f; sN[i] = 0.f; sP[i] = 0.f; sM[i] = 0.f;
    }
    __syncthreads();

    const int lane = tid & 31;
    const int wv   = tid >> 5;
    const int w    = blockIdx.x * CT + lane;           // global column
    const int r0   = blockIdx.y * ROWS_PER_BLOCK + wv * ROWS_PER_WAVE;

    float Sacc[NC];
#pragma unroll
    for (int c = 0; c < NC; c++) Sacc[c] = 0.f;

    for (int i = 0; i < ROWS_PER_WAVE; i++) {
        const int r = r0 + i;
        const int b = r >> 9;          // r / H
        const int h = r & (HH - 1);    // r % H
        const float* px = pred + (size_t)b * CHW + (size_t)h * WW + w;

        // prefetch next row's first class line (global_prefetch_b8)
        if (i + 1 < ROWS_PER_WAVE) __builtin_prefetch(px + WW, 0, 1);

        float x[NC];
#pragma unroll
        for (int c = 0; c < NC; c++) x[c] = px[(size_t)c * HW];

        // max + first-argmax (jnp.argmax keeps first max -> strict >)
        float m = x[0]; int am = 0;
#pragma unroll
        for (int c = 1; c < NC; c++) { if (x[c] > m) { m = x[c]; am = c; } }

        const int t = target[(size_t)r * WW + w];

        float sum = 0.f, et = 0.f, xt = 0.f;
#pragma unroll
        for (int c = 0; c < NC; c++) {
            const float xv = x[c];
            const float e  = __expf(xv - m);
            x[c] = e;                  // reuse register array for exp values
            sum += e;
            if (c == t) { et = e; xt = xv; }
        }
        const float rinv = __builtin_amdgcn_rcpf(sum);   // v_rcp_f32
        const float nll  = m + __logf(sum) - xt;         // logsumexp - x[target]

#pragma unroll
        for (int c = 0; c < NC; c++) Sacc[c] += x[c] * rinv;   // softmax prob sums

        atomicAdd(&sI[t  * CT + lane], et * rinv);               // inter (tversky)
        atomicAdd(&sT[t  * CT + lane], 1.0f);                    // onehot count
        atomicAdd(&sN[t  * CT + lane], nll);                     // per-class nll sum
        atomicAdd(&sP[am * CT + lane], 1.0f);                    // argmax count
        atomicAdd(&sM[t  * CT + lane], (am == t) ? 1.0f : 0.0f); // match count
    }

    // fold per-lane register accumulators into LDS
#pragma unroll
    for (int c = 0; c < NC; c++) atomicAdd(&sS[c * CT + lane], Sacc[c]);
    __syncthreads();

    // flush block tile to global accumulators (one flush per block)
    for (int i = tid; i < NC * CT; i += P1_THREADS) {
        const int c  = i / CT;
        const int wl = i - c * CT;
        const int g  = c * WW + blockIdx.x * CT + wl;
        atomicAdd(&ws[WS_S + g], sS[i]);
        atomicAdd(&ws[WS_I + g], sI[i]);
        atomicAdd(&ws[WS_T + g], sT[i]);
        atomicAdd(&ws[WS_N + g], sN[i]);
        atomicAdd(&ws[WS_P + g], sP[i]);
        atomicAdd(&ws[WS_M + g], sM[i]);
    }
}

__global__ __launch_bounds__(256) void tcedw_final(const float* __restrict__ ws,
                                                   float* __restrict__ out)
{
    __shared__ float tc[NC], ns[NC], pc[NC], mc[NC];
    __shared__ float wpart[FW_WAVES];
    __shared__ float ce2[2];

    const int tid = threadIdx.x;
    if (tid < NC) { tc[tid] = 0.f; ns[tid] = 0.f; pc[tid] = 0.f; mc[tid] = 0.f; }
    __syncthreads();

    // per-class row sums of T / nll / pcnt / match
    for (int i = tid; i < NCW; i += 256) {
        const int c = i >> 9;  // / WW
        atomicAdd(&tc[c], ws[WS_T + i]);
        atomicAdd(&ns[c], ws[WS_N + i]);
        atomicAdd(&pc[c], ws[WS_P + i]);
        atomicAdd(&mc[c], ws[WS_M + i]);
    }
    __syncthreads();

    if (tid == 0) {
        float cn = 0.f, cd = 0.f;
        for (int c = 0; c < NC; c++) {
            const float wc = 1.0f - 2.0f * (mc[c] + EPSF) / (pc[c] + tc[c] + EPSF);
            cn += wc * ns[c];   // sum w[t]*nll
            cd += wc * tc[c];   // sum w[t]
        }
        ce2[0] = cn; ce2[1] = cd;
    }

    // ---- Tversky mean over 21*512 = 10752 terms via WMMA f32 16x16x4 ----
    // A = ones(16x4): D[m][n] = sum_k B[k][n] + C[m][n]; each WMMA folds the
    // 64 lane-values in B's two VGPRs into the f32 accumulator exactly.
    const int lane = tid & 31;
    const int wv   = tid >> 5;

    v2f aones = {1.0f, 1.0f};
    v8f acc = {};
    for (int it = 0; it < 21; it++) {          // 21 * 8 waves * 64 = 10752 exactly
        const int base = (it * FW_WAVES + wv) * 64;
        const int e0 = base + lane;
        const int e1 = base + 32 + lane;
        const float r0v = 2.0f * ws[WS_I + e0] / (ws[WS_S + e0] + ws[WS_T + e0]);
        const float r1v = 2.0f * ws[WS_I + e1] / (ws[WS_S + e1] + ws[WS_T + e1]);
        v2f bv = {r0v, r1v};
        acc = __builtin_amdgcn_wmma_f32_16x16x4_f32(
            /*neg_a=*/false, aones, /*neg_b=*/false, bv,
            /*c_mod=*/(short)0, acc, /*reuse_a=*/false, /*reuse_b=*/false);
    }
    // VGPR0 holds rows M=0 (lanes 0-15) and M=8 (lanes 16-31); rows identical
    // with A=ones, so full 32-lane sum = 2x total.
    float s = acc[0];
#pragma unroll
    for (int off = 16; off > 0; off >>= 1) s += __shfl_xor(s, off, 32);
    s *= 0.5f;
    if (lane == 0) wpart[wv] = s;
    __syncthreads();

    if (tid == 0) {
        float tsum = 0.f;
        for (int i = 0; i < FW_WAVES; i++) tsum += wpart[i];
        const float tv = 1.0f - tsum / (float)NCW;
        const float ce = ce2[0] / ce2[1];
        out[0] = 0.5f * ce + 0.5f * tv;
    }
}

extern "C" void kernel_launch(void* const* d_in, const int* in_sizes, int n_in,
                              void* d_out, int out_size, void* d_ws, size_t ws_size,
                              hipStream_t stream) {
    const float* pred   = (const float*)d_in[0];
    const int*   target = (const int*)d_in[1];
    float* ws  = (float*)d_ws;
    float* out = (float*)d_out;

    hipMemsetAsync(d_ws, 0, (size_t)WS_FLOATS * sizeof(float), stream);

    dim3 grid(NTILE, ROWCH);   // 16 x 16 = 256 blocks
    tcedw_pass1<<<grid, P1_THREADS, 0, stream>>>(pred, target, ws);
    tcedw_final<<<1, 256, 0, stream>>>(ws, out);
}